// FilterbankAttention_79182017069388
// MI455X (gfx1250) — compile-verified
//
#include <hip/hip_runtime.h>

typedef float v2f __attribute__((ext_vector_type(2)));
typedef float v8f __attribute__((ext_vector_type(8)));

#define BATCH 1024
#define ADIM  128   // A == B == 128
#define NDIM  64
#define HDIM  1024

// ---------------------------------------------------------------------------
// Setup kernel: one block. Computes the 5 params from h[0] @ W_read_w^T + b,
// then builds the normalized filterbanks into d_ws:
//   Fy (64x128, gamma folded in), Fx (64x128).
// ---------------------------------------------------------------------------
__global__ __launch_bounds__(256)
void fb_setup_kernel(const float* __restrict__ h,
                     const float* __restrict__ Ww,
                     const float* __restrict__ Wb,
                     float* __restrict__ Fy,
                     float* __restrict__ Fx)
{
    __shared__ float red[256];
    __shared__ float sc[8];
    const int t = threadIdx.x;

    // params[j] = h[0,:] . Ww[j,:] + Wb[j]
    for (int j = 0; j < 5; ++j) {
        float p = 0.f;
        for (int i = t; i < HDIM; i += 256) p += h[i] * Ww[j * HDIM + i];
        red[t] = p; __syncthreads();
        for (int s = 128; s > 0; s >>= 1) {
            if (t < s) red[t] += red[t + s];
            __syncthreads();
        }
        if (t == 0) sc[j] = red[0] + Wb[j];
        __syncthreads();
    }

    if (t == 0) {
        float gtX = sc[0], gtY = sc[1], log_var = sc[2], log_dt = sc[3], lg = sc[4];
        float var = __expf(log_var + 1e-8f);
        sc[0] = (float)(ADIM + 1) * (gtX + 1.f) * 0.5f;                   // g_X
        sc[1] = (float)(ADIM + 1) * (gtY + 1.f) * 0.5f;                   // g_Y
        sc[2] = -1.f / (2.f * var);                                       // -1/(2 var)
        sc[3] = __expf(log_dt) * (float)(ADIM - 1) / (float)(NDIM - 1);   // d
        sc[4] = __expf(lg);                                               // gamma
    }
    __syncthreads();
    const float gX = sc[0], gY = sc[1], ninv2var = sc[2], d = sc[3], gamma = sc[4];

    // 8192 elements per matrix, 32 per thread, kept in registers.
    float fy[32], fx[32];
    float sx = 0.f, sy = 0.f;
    #pragma unroll
    for (int q = 0; q < 32; ++q) {
        int e = t + q * 256;
        int n = e >> 7;          // filter row 0..63
        int c = e & 127;         // pixel col 0..127
        float idx = (float)n - 32.5f;   // n - N/2 - 0.5
        float dxv = (float)c - (gX + idx * d);
        float dyv = (float)c - (gY + idx * d);
        float vx = __expf(dxv * dxv * ninv2var);
        float vy = __expf(dyv * dyv * ninv2var);
        fx[q] = vx; fy[q] = vy;
        sx += vx; sy += vy;
    }
    // global-sum normalization (as in the reference)
    red[t] = sx; __syncthreads();
    for (int s = 128; s > 0; s >>= 1) { if (t < s) red[t] += red[t + s]; __syncthreads(); }
    const float sumX = red[0]; __syncthreads();
    red[t] = sy; __syncthreads();
    for (int s = 128; s > 0; s >>= 1) { if (t < s) red[t] += red[t + s]; __syncthreads(); }
    const float sumY = red[0]; __syncthreads();

    const float rx = 1.f / sumX;
    const float ry = gamma / sumY;   // fold gamma into Fy
    #pragma unroll
    for (int q = 0; q < 32; ++q) {
        int e = t + q * 256;
        Fx[e] = fx[q] * rx;
        Fy[e] = fy[q] * ry;
    }
}

// ---------------------------------------------------------------------------
// Main kernel: one block per (image, {x|x_hat}). 4 waves; wave w owns output
// rows [16w, 16w+16). Two chained GEMMs on v_wmma_f32_16x16x4_f32.
// Loop nest: K-step outer, output tiles inner -> one A-fragment load is
// amortized over 8 (GEMM1) / 4 (GEMM2) WMMAs, and 8/4 independent
// accumulator chains keep the matrix pipe full.
// ---------------------------------------------------------------------------
__global__ __launch_bounds__(128)
void fb_filter_kernel(const float* __restrict__ x,
                      const float* __restrict__ x_hat,
                      const float* __restrict__ Fy,
                      const float* __restrict__ Fx,
                      float* __restrict__ out)
{
    __shared__ float Tsh[NDIM * ADIM];   // 64x128 f32 = 32 KB

    const int lane  = threadIdx.x & 31;
    const int wave  = threadIdx.x >> 5;        // 0..3
    const int img   = blockIdx.x >> 1;
    const int which = blockIdx.x & 1;

    const float* __restrict__ z =
        (which ? x_hat : x) + (size_t)img * (ADIM * ADIM);

    const int mloc  = lane & 15;               // tile row (A) / tile col (B)
    const int koff  = (lane >> 4) << 1;        // lanes 0-15 -> K{0,1}, 16-31 -> K{2,3}
    const int rbase = (lane >> 4) << 3;        // C/D: lanes 16-31 hold M+8

    // ---- GEMM1: T[16w+.., :] = Fy_rows @ z --------------------------------
    {
        const float* fyrow = Fy + (wave * 16 + mloc) * ADIM;
        v8f acc[8];
        #pragma unroll
        for (int bt = 0; bt < 8; ++bt) acc[bt] = (v8f){};

        for (int s = 0; s < 32; ++s) {         // K = 128 in steps of 4
            const int k = s * 4 + koff;
            const v2f a = *(const v2f*)(fyrow + k);          // A[m][k..k+1]
            const float* z0 = z + (size_t)k * ADIM + mloc;   // B rows k, k+1
            #pragma unroll
            for (int bt = 0; bt < 8; ++bt) {
                v2f b;
                b.x = z0[bt * 16];             // B[k][n]   (coalesced across lanes)
                b.y = z0[bt * 16 + ADIM];      // B[k+1][n]
                acc[bt] = __builtin_amdgcn_wmma_f32_16x16x4_f32(
                            false, a, false, b, (short)0, acc[bt], false, false);
            }
        }
        // scatter C/D layout into LDS row-major T
        #pragma unroll
        for (int bt = 0; bt < 8; ++bt)
            #pragma unroll
            for (int r = 0; r < 8; ++r)
                Tsh[(wave * 16 + rbase + r) * ADIM + bt * 16 + mloc] = acc[bt][r];
    }
    __syncthreads();

    // ---- GEMM2: Out[16w+.., :] = T_w @ Fx^T -------------------------------
    {
        const float* trow  = &Tsh[(wave * 16 + mloc) * ADIM];
        const size_t obase = (size_t)img * (2 * NDIM * NDIM)
                           + (size_t)which * (NDIM * NDIM);
        v8f oacc[4];
        #pragma unroll
        for (int mt = 0; mt < 4; ++mt) oacc[mt] = (v8f){};

        for (int s = 0; s < 32; ++s) {
            const int k = s * 4 + koff;
            const v2f a = *(const v2f*)(trow + k);           // ds_load_b64
            #pragma unroll
            for (int mt = 0; mt < 4; ++mt) {
                // B[k][m] = Fx[m][k..k+1] (contiguous -> b64 load)
                const v2f b = *(const v2f*)(Fx + (mt * 16 + mloc) * ADIM + k);
                oacc[mt] = __builtin_amdgcn_wmma_f32_16x16x4_f32(
                             false, a, false, b, (short)0, oacc[mt], false, false);
            }
        }
        #pragma unroll
        for (int mt = 0; mt < 4; ++mt)
            #pragma unroll
            for (int r = 0; r < 8; ++r) {
                const int nrow = wave * 16 + rbase + r;
                out[obase + (size_t)nrow * NDIM + mt * 16 + mloc] = oacc[mt][r];
            }
    }
}

extern "C" void kernel_launch(void* const* d_in, const int* in_sizes, int n_in,
                              void* d_out, int out_size, void* d_ws, size_t ws_size,
                              hipStream_t stream) {
    (void)in_sizes; (void)n_in; (void)out_size; (void)ws_size;
    const float* x  = (const float*)d_in[0];
    const float* xh = (const float*)d_in[1];
    const float* h  = (const float*)d_in[2];
    const float* Ww = (const float*)d_in[3];
    const float* Wb = (const float*)d_in[4];
    float* out = (float*)d_out;

    float* Fy = (float*)d_ws;            // 64*128 floats (gamma folded)
    float* Fx = Fy + NDIM * ADIM;        // 64*128 floats

    fb_setup_kernel<<<1, 256, 0, stream>>>(h, Ww, Wb, Fy, Fx);
    fb_filter_kernel<<<BATCH * 2, 128, 0, stream>>>(x, xh, Fy, Fx, out);
}